// StudentModel_62551903699323
// MI455X (gfx1250) — compile-verified
//
#include <hip/hip_runtime.h>

// CDNA5 / gfx1250, wave32 only.
typedef float v2f __attribute__((ext_vector_type(2)));
typedef float v8f __attribute__((ext_vector_type(8)));

// D = A(16x4 f32) * B(4x16 f32) + C(16x16 f32), full fp32 matrix pipe.
#define WMMA_F32(a, b, c) \
  __builtin_amdgcn_wmma_f32_16x16x4_f32(false, (a), false, (b), (short)0, (c), false, false)

// ---------------------------------------------------------------------------
// Generic batched fp32 WMMA GEMM: C[z] = act(A[z] @ B[z] + bias)
// One wave (32 threads) computes one 16x16 C tile; grid=(N/16, M/16, batch).
// A-frag  (16x4): lane m = lane&15, K pair = 2*(lane>>4) -> contiguous v2f.
// B-frag  (4x16): lane n = lane&15, rows K = 2*(lane>>4)+{0,1}.
// C/D: vgpr r holds M = r + 8*(lane>>4), N = lane&15.
// ---------------------------------------------------------------------------
__global__ __launch_bounds__(32)
void gemm_f32_wmma(const float* __restrict__ A, int lda, long sA,
                   const float* __restrict__ B, int ldb, long sB,
                   float* __restrict__ C, int ldc, long sC,
                   const float* __restrict__ bias, int act, int K) {
  const int n0 = blockIdx.x * 16;
  const int m0 = blockIdx.y * 16;
  const long z = blockIdx.z;
  A += z * sA; B += z * sB; C += z * sC;

  const int lane = threadIdx.x & 31;
  const int ln = lane & 15;
  const int hi = lane >> 4;

  v8f acc = {0.f, 0.f, 0.f, 0.f, 0.f, 0.f, 0.f, 0.f};
  const float* arow = A + (long)(m0 + ln) * lda;

#pragma unroll 4
  for (int k = 0; k < K; k += 4) {
    const int k0 = k + 2 * hi;
    v2f a = *(const v2f*)(arow + k0);        // 8B aligned: lda even, k0 even
    v2f b;
    b.x = B[(long)k0 * ldb + n0 + ln];
    b.y = B[(long)(k0 + 1) * ldb + n0 + ln];
    acc = WMMA_F32(a, b, acc);
  }

  const float bv = bias ? bias[n0 + ln] : 0.f;
#pragma unroll
  for (int r = 0; r < 8; ++r) {
    float v = acc[r] + bv;
    if (act == 1) v = fmaxf(v, 0.f);
    else if (act == 2) v = tanhf(v);
    C[(long)(m0 + 8 * hi + r) * ldc + n0 + ln] = v;
  }
}

// ---------------------------------------------------------------------------
// Fused edge-construction kernel. One workgroup per (b, i):
//   h1[j,k] = relu(P[b,i,k] + Q[b,j,k] + eb1[k])        (built in LDS, 32-K chunks)
//   h2[j,n] = relu(sum_k h1[j,k]*ew2[k,n] + eb2[n])     (WMMA f32)
//   adj[b,i,j] = (sum_n h2[j,n]*ew3[n] + eb3) * m_i * m_j
// Never materializes the 134 MB h1 tensor in HBM.
// 256 threads = 8 waves; wave w owns j-strip [16w, 16w+16), all 8 n-tiles.
// ---------------------------------------------------------------------------
#define KC 32
__global__ __launch_bounds__(256)
void edge_adj_kernel(const float* __restrict__ P, const float* __restrict__ Q,
                     const float* __restrict__ eb1, const float* __restrict__ ew2,
                     const float* __restrict__ eb2, const float* __restrict__ ew3,
                     const float* __restrict__ eb3, const int* __restrict__ mask,
                     float* __restrict__ adj) {
  const int i = blockIdx.x;   // token i (0..127)
  const int b = blockIdx.y;   // batch   (0..3)

  __shared__ float h1s[128 * 36];  // 128 rows(j) x KC, pad->36 (bank-conflict free)
  __shared__ float bsh[KC * 132];  // KC rows(k) x 128(n), pad->132

  const int t = threadIdx.x;
  const int wave = t >> 5;
  const int lane = t & 31;
  const int ln = lane & 15;
  const int hi = lane >> 4;

  v8f acc[8];
#pragma unroll
  for (int nt = 0; nt < 8; ++nt)
    acc[nt] = (v8f){0.f, 0.f, 0.f, 0.f, 0.f, 0.f, 0.f, 0.f};

  const float* Pi = P + (long)(b * 128 + i) * 512;

  for (int kc = 0; kc < 512; kc += KC) {
    // stage ew2[kc:kc+KC, 0:128] -> bsh
    {
      const int r0 = t >> 5;            // 0..7
      const int c4 = (t & 31) * 4;      // 0..124
#pragma unroll
      for (int p = 0; p < KC / 8; ++p) {
        const int r = r0 + p * 8;
        *(float4*)&bsh[r * 132 + c4] =
            *(const float4*)&ew2[(long)(kc + r) * 128 + c4];
      }
    }
    // stage h1 chunk: row j = t/2, half (t&1) covers 16 floats
    {
      const int j = t >> 1;
      const int h = (t & 1) * (KC / 2);
      const float* q = Q + (long)(b * 128 + j) * 512 + kc + h;
      const float* p = Pi + kc + h;
      const float* e = eb1 + kc + h;
      float* dst = &h1s[j * 36 + h];
#pragma unroll
      for (int c = 0; c < KC / 2; c += 4) {
        float4 qa = *(const float4*)(q + c);
        float4 pa = *(const float4*)(p + c);
        float4 ea = *(const float4*)(e + c);
        float4 v;
        v.x = fmaxf(qa.x + pa.x + ea.x, 0.f);
        v.y = fmaxf(qa.y + pa.y + ea.y, 0.f);
        v.z = fmaxf(qa.z + pa.z + ea.z, 0.f);
        v.w = fmaxf(qa.w + pa.w + ea.w, 0.f);
        *(float4*)(dst + c) = v;
      }
    }
    __syncthreads();

#pragma unroll
    for (int k4 = 0; k4 < KC; k4 += 4) {
      v2f a = *(const v2f*)&h1s[(wave * 16 + ln) * 36 + k4 + 2 * hi];
#pragma unroll
      for (int nt = 0; nt < 8; ++nt) {
        v2f bb;
        bb.x = bsh[(k4 + 2 * hi) * 132 + nt * 16 + ln];
        bb.y = bsh[(k4 + 2 * hi + 1) * 132 + nt * 16 + ln];
        acc[nt] = WMMA_F32(a, bb, acc[nt]);
      }
    }
    __syncthreads();
  }

  // epilogue: h2 = relu(acc + eb2); s[r] = sum_n h2 * ew3 (partial over n%16==ln)
  float s[8] = {0.f, 0.f, 0.f, 0.f, 0.f, 0.f, 0.f, 0.f};
#pragma unroll
  for (int nt = 0; nt < 8; ++nt) {
    const int n = nt * 16 + ln;
    const float w3 = ew3[n];
    const float b2 = eb2[n];
#pragma unroll
    for (int r = 0; r < 8; ++r)
      s[r] += fmaxf(acc[nt][r] + b2, 0.f) * w3;
  }
  // reduce over the 16 lanes of each half-wave (xor 1,2,4,8 stays in-half)
#pragma unroll
  for (int off = 1; off < 16; off <<= 1) {
#pragma unroll
    for (int r = 0; r < 8; ++r) s[r] += __shfl_xor(s[r], off, 32);
  }

  if (ln == 0) {
    const float mi = (float)mask[b * 128 + i];
    const float e3 = eb3[0];
#pragma unroll
    for (int r = 0; r < 8; ++r) {
      const int j = wave * 16 + 8 * hi + r;   // M = r + 8*hi in this j-strip
      const float mj = (float)mask[b * 128 + j];
      adj[(long)(b * 128 + i) * 128 + j] = (s[r] + e3) * mi * mj;
    }
  }
}

// ---------------------------------------------------------------------------
// Launch: B=4, T=128, D=768, EH1=512, EH2=128, FH1=512, FOUT=256, GH1=256, GOUT=64
// ---------------------------------------------------------------------------
extern "C" void kernel_launch(void* const* d_in, const int* in_sizes, int n_in,
                              void* d_out, int out_size, void* d_ws, size_t ws_size,
                              hipStream_t stream) {
  const float* features = (const float*)d_in[0];   // [4,128,768]
  const int*   amask    = (const int*)d_in[1];     // [4,128]
  const float* ew1 = (const float*)d_in[2];        // [1536,512]
  const float* eb1 = (const float*)d_in[3];        // [512]
  const float* ew2 = (const float*)d_in[4];        // [512,128]
  const float* eb2 = (const float*)d_in[5];        // [128]
  const float* ew3 = (const float*)d_in[6];        // [128,1]
  const float* eb3 = (const float*)d_in[7];        // [1]
  const float* fw1 = (const float*)d_in[8];        // [768,512]
  const float* fb1 = (const float*)d_in[9];
  const float* fw2 = (const float*)d_in[10];       // [512,256]
  const float* fb2 = (const float*)d_in[11];
  const float* gw1 = (const float*)d_in[12];       // [256,256]
  const float* gb1 = (const float*)d_in[13];
  const float* gw2 = (const float*)d_in[14];       // [256,64]
  const float* gb2 = (const float*)d_in[15];
  float* out = (float*)d_out;                      // [4,128,64]

  // workspace layout (floats); total ~4.9 MB
  float* ws  = (float*)d_ws;
  float* P   = ws;               // [512,512]
  float* Qm  = P + 262144;       // [512,512]
  float* F1  = Qm + 262144;      // [512,512]
  float* F   = F1 + 262144;      // [512,256]
  float* XW1 = F + 131072;       // [512,256]
  float* adj = XW1 + 131072;     // [4,128,128]
  float* G   = adj + 65536;      // [512,256]
  float* GW2 = G + 131072;       // [512,64]

  const dim3 wv(32);

  // P = X @ ew1[:768,:]   (no bias; eb1 folded into fused edge kernel)
  gemm_f32_wmma<<<dim3(32, 32, 1), wv, 0, stream>>>(
      features, 768, 0, ew1, 512, 0, P, 512, 0, nullptr, 0, 768);
  // Q = X @ ew1[768:,:]
  gemm_f32_wmma<<<dim3(32, 32, 1), wv, 0, stream>>>(
      features, 768, 0, ew1 + 768 * 512, 512, 0, Qm, 512, 0, nullptr, 0, 768);
  // F1 = relu(X @ fw1 + fb1)
  gemm_f32_wmma<<<dim3(32, 32, 1), wv, 0, stream>>>(
      features, 768, 0, fw1, 512, 0, F1, 512, 0, fb1, 1, 768);
  // F = F1 @ fw2 + fb2
  gemm_f32_wmma<<<dim3(16, 32, 1), wv, 0, stream>>>(
      F1, 512, 0, fw2, 256, 0, F, 256, 0, fb2, 0, 512);
  // XW1 = F @ gw1 + gb1
  gemm_f32_wmma<<<dim3(16, 32, 1), wv, 0, stream>>>(
      F, 256, 0, gw1, 256, 0, XW1, 256, 0, gb1, 0, 256);

  // adj[b,i,j] = masked( edgeMLP(concat(x_i,x_j)) )  -- fused, h1 never hits HBM
  edge_adj_kernel<<<dim3(128, 4), dim3(256), 0, stream>>>(
      P, Qm, eb1, ew2, eb2, ew3, eb3, amask, adj);

  // G = relu(adj @ XW1)         (batched over b)
  gemm_f32_wmma<<<dim3(16, 8, 4), wv, 0, stream>>>(
      adj, 128, 16384, XW1, 256, 32768, G, 256, 32768, nullptr, 1, 128);
  // GW2 = G @ gw2 + gb2
  gemm_f32_wmma<<<dim3(4, 32, 1), wv, 0, stream>>>(
      G, 256, 0, gw2, 64, 0, GW2, 64, 0, gb2, 0, 256);
  // out = tanh(adj @ GW2)       (batched over b)
  gemm_f32_wmma<<<dim3(4, 8, 4), wv, 0, stream>>>(
      adj, 128, 16384, GW2, 64, 8192, out, 64, 8192, nullptr, 2, 128);
}